// BNB8bitEmbedding_42992622633476
// MI455X (gfx1250) — compile-verified
//
#include <hip/hip_runtime.h>
#include <cstdint>
#include <cstddef>

// ---------------------------------------------------------------------------
// Fused bnb-8bit dequant-on-gather embedding:
//   out[i, :] = code[ q_idx[token_i, :] ] * absmax[token_i >> 2]
//
// Simplifications (derived in round 0):
//   * chunk*64 + local == token        -> q_idx row address is token*1024
//   * per-row block index is constant  -> scale = absmax[token >> 2]
//
// Memory-bound: ~256 MB HBM traffic -> ~11 us at 23.3 TB/s.
// Strategy: b128 gathers + b128 NT stores (output is write-once; working set
// 256 MB > 192 MB L2, so keep L2 for the reused q_idx gather stream),
// 256-entry codebook staged into LDS via CDNA5 async-to-LDS, per-lane ds
// lookups, scalar (SGPR) token/scale addressing.
// ---------------------------------------------------------------------------

#define EMBED_DIM        1024
#define THREADS          256      // 8 waves (wave32); one int4/float4 per lane per row
#define TOKENS_PER_BLOCK 8

typedef float v4f __attribute__((ext_vector_type(4)));

__global__ __launch_bounds__(THREADS)
void bnb8bit_embed_gather_kernel(const int*   __restrict__ x,       // [n_tokens]
                                 const int*   __restrict__ q_idx,   // [128000*1024] codes (0..255)
                                 const float* __restrict__ absmax,  // [32000] (flat view)
                                 const float* __restrict__ code,    // [256]
                                 float*       __restrict__ out,     // [n_tokens*1024]
                                 int n_tokens)
{
    __shared__ float lds_code[256];
    const int tid = threadIdx.x;

    // ---- Stage the 256-entry codebook into LDS via the CDNA5 async path.
    // One b32 per lane; low 32 bits of the flat shared-pointer are the LDS
    // byte offset (ISA 10.2 aperture mapping), which is what VDST carries.
    {
        unsigned           lds_off = (unsigned)(uintptr_t)(&lds_code[tid]);
        unsigned long long gaddr   = (unsigned long long)(uintptr_t)(code + tid);
        asm volatile("global_load_async_to_lds_b32 %0, %1, off"
                     :: "v"(lds_off), "v"(gaddr)
                     : "memory");
        asm volatile("s_wait_asynccnt 0" ::: "memory");
    }
    __syncthreads();   // all waves' async fills visible before any lookup

    const int token_base = blockIdx.x * TOKENS_PER_BLOCK;

    #pragma unroll
    for (int t = 0; t < TOKENS_PER_BLOCK; ++t) {
        const int ti = token_base + t;
        if (ti >= n_tokens) break;

        const int   token = x[ti];                  // uniform across block -> SMEM load
        const float scale = absmax[token >> 2];     // one scalar per row

        // Prefetch next row's 64B line for this lane (gfx1250 global_prefetch_b8).
        if (t + 1 < TOKENS_PER_BLOCK && ti + 1 < n_tokens) {
            const int ntoken = x[ti + 1];
            __builtin_prefetch(q_idx + (size_t)ntoken * EMBED_DIM + (size_t)tid * 4, 0, 0);
        }

        // b128 gather of 4 codes, LDS lookups, b128 non-temporal store of 4 floats.
        const int4 q = ((const int4* __restrict__)(q_idx + (size_t)token * EMBED_DIM))[tid];

        v4f r;
        r.x = lds_code[q.x & 0xFF] * scale;
        r.y = lds_code[q.y & 0xFF] * scale;
        r.z = lds_code[q.z & 0xFF] * scale;
        r.w = lds_code[q.w & 0xFF] * scale;

        v4f* orow = (v4f*)(out + (size_t)ti * EMBED_DIM);
        __builtin_nontemporal_store(r, orow + tid);   // th:NT -> don't thrash L2
    }
}

extern "C" void kernel_launch(void* const* d_in, const int* in_sizes, int n_in,
                              void* d_out, int out_size, void* d_ws, size_t ws_size,
                              hipStream_t stream)
{
    // setup_inputs() order: x [8*4096] i32, q_idx [2000*64*1024] i32,
    //                       absmax [2000*16] f32, code [256] f32
    const int*   x      = (const int*)  d_in[0];
    const int*   q_idx  = (const int*)  d_in[1];
    const float* absmax = (const float*)d_in[2];
    const float* code   = (const float*)d_in[3];
    float*       out    = (float*)d_out;

    const int n_tokens = in_sizes[0];   // 32768
    const int grid     = (n_tokens + TOKENS_PER_BLOCK - 1) / TOKENS_PER_BLOCK;

    bnb8bit_embed_gather_kernel<<<grid, THREADS, 0, stream>>>(
        x, q_idx, absmax, code, out, n_tokens);
}